// TestSpatialTransform_19095424598596
// MI455X (gfx1250) — compile-verified
//
#include <hip/hip_runtime.h>
#include <stdint.h>

// ---------------------------------------------------------------------------
// TestSpatialTransform: the reference's trilinear sample coordinates are exact
// integers (arange(160)+16), so both outputs are pure center crops:
//   out0 = data[:, :, 16:176, 16:176, 16:176]   (4,2,160,160,160) f32
//   out1 = seg [:, :, 16:176, 16:176, 16:176]   (4,2,160,160,160) f32
// Pure data movement (~524 MB total traffic, ~25 us at 23.3 TB/s) -> drive it
// with the CDNA5 Tensor Data Mover (TDM): DMA global -> LDS -> global.
// ---------------------------------------------------------------------------

typedef __attribute__((ext_vector_type(4))) unsigned int v4u;
typedef __attribute__((ext_vector_type(4))) int          v4i;
typedef __attribute__((ext_vector_type(8))) int          v8i;

#if __has_builtin(__builtin_amdgcn_tensor_load_to_lds)
#define HAS_TDM 1
#if __has_include(<hip/amd_detail/amd_gfx1250_TDM.h>)
// amdgpu-toolchain (clang-23 / therock-10.0 headers): 6-arg form
#define TDM_LOAD(g0, g1, z4, z8)  __builtin_amdgcn_tensor_load_to_lds((g0), (g1), (z4), (z4), (z8), 0)
#define TDM_STORE(g0, g1, z4, z8) __builtin_amdgcn_tensor_store_from_lds((g0), (g1), (z4), (z4), (z8), 0)
#else
// ROCm 7.2 (clang-22): 5-arg form
#define TDM_LOAD(g0, g1, z4, z8)  __builtin_amdgcn_tensor_load_to_lds((g0), (g1), (z4), (z4), 0)
#define TDM_STORE(g0, g1, z4, z8) __builtin_amdgcn_tensor_store_from_lds((g0), (g1), (z4), (z4), 0)
#endif
#else
#define HAS_TDM 0
#endif

// ---- D# descriptor packing (CDNA5 ISA ch. 8, "Tensor DMA Descriptor") ------

// Group 0 (128b): [1:0]=count=1 (valid, user mode), [63:32]=lds_addr (bytes),
// [120:64]=global_addr (bytes), [127:126]=type=2 ("image").
__device__ __forceinline__ v4u tdm_group0(uint32_t lds_addr, uint64_t gaddr) {
  v4u g;
  g.x = 1u;                                         // count=1, is_restore=0, gather off
  g.y = lds_addr;                                   // LDS byte address
  g.z = (uint32_t)(gaddr & 0xFFFFFFFFu);            // global_addr[31:0]
  g.w = (uint32_t)((gaddr >> 32) & 0x01FFFFFFu)     // global_addr[56:32]
      | (2u << 30);                                 // type = 2
  return g;
}

// Group 1 (256b): [17:16]=data_size (2 -> 4 bytes), [79:48]=tensor_dim0,
// [111:80]=tensor_dim1, [127:112]=tile_dim0, [143:128]=tile_dim1,
// [159:144]=tile_dim2 (0 = 2D tile), [207:160]=tensor_dim0_stride,
// [255:208]=tensor_dim1_stride (unused for 2D tile).
__device__ __forceinline__ v8i tdm_group1(uint32_t tdim0, uint32_t tdim1,
                                          uint32_t tile0, uint32_t tile1,
                                          uint64_t stride0) {
  v8i g;
  g[0] = (int)(2u << 16);                                        // data_size = 4B
  g[1] = (int)((tdim0 & 0xFFFFu) << 16);                         // tensor_dim0[15:0]
  g[2] = (int)((tdim0 >> 16) | ((tdim1 & 0xFFFFu) << 16));       // dim0 hi | dim1 lo
  g[3] = (int)((tdim1 >> 16) | ((tile0 & 0xFFFFu) << 16));       // dim1 hi | tile_dim0
  g[4] = (int)(tile1 & 0xFFFFu);                                 // tile_dim1 | tile_dim2=0
  g[5] = (int)(uint32_t)(stride0 & 0xFFFFFFFFu);                 // dim0_stride[31:0]
  g[6] = (int)(uint32_t)((stride0 >> 32) & 0xFFFFu);             // dim0_stride[47:32]
  g[7] = 0;                                                      // dim1_stride hi
  return g;
}

#if HAS_TDM
// One wave (32 threads) per workgroup; each workgroup DMAs one 160x32 f32
// sub-slice: global (row stride 192) -> LDS -> global (row stride 160).
// grid = (5 y-chunks, 160 z-slices, 2 tensors * 8 volumes).
__global__ __launch_bounds__(32) void tdm_crop_copy(const float* __restrict__ data,
                                                    const float* __restrict__ seg,
                                                    float* __restrict__ out) {
  __shared__ float tile[160 * 32];                  // 20480 B -> 16 WGs/WGP in 320KB

  const uint32_t chunk = blockIdx.x;                // y-chunk of 32 rows   [0,5)
  const uint32_t z     = blockIdx.y;                // output depth slice   [0,160)
  const uint32_t bcid  = blockIdx.z;                // (t<<3) | bc          [0,16)
  const uint32_t t     = bcid >> 3;                 // 0 = data, 1 = seg
  const uint32_t bc    = bcid & 7u;                 // fused batch*channel

  const float* srcb = (t == 0u) ? data : seg;
  float*       dstb = out + (size_t)t * (8ull * 160u * 160u * 160u);

  const uint32_t y0 = chunk << 5;                   // chunk * 32
  const uint64_t src_elem =
      (((uint64_t)(bc * 192u + (z + 16u)) * 192u + (y0 + 16u)) * 192u) + 16u;
  const uint64_t dst_elem =
      ((uint64_t)(bc * 160u + z) * 160u + y0) * 160u;

  const uint64_t src_addr = (uint64_t)(uintptr_t)srcb + src_elem * 4ull;
  const uint64_t dst_addr = (uint64_t)(uintptr_t)dstb + dst_elem * 4ull;
  // Wave-relative LDS byte offset = low 32 bits of the flat shared address
  // (ISA: in the LDS aperture the low 32 address bits are the LDS address).
  const uint32_t lds_addr = (uint32_t)(uintptr_t)&tile[0];

  const v4u g0_ld = tdm_group0(lds_addr, src_addr);
  const v8i g1_ld = tdm_group1(/*tensor_dim0*/ 4096u, /*tensor_dim1*/ 4096u,
                               /*tile0*/ 160u, /*tile1*/ 32u, /*stride0*/ 192u);
  const v4u g0_st = tdm_group0(lds_addr, dst_addr);
  const v8i g1_st = tdm_group1(4096u, 4096u, 160u, 32u, /*stride0*/ 160u);
  const v4i z4 = {0, 0, 0, 0};
  const v8i z8 = {0, 0, 0, 0, 0, 0, 0, 0};
  (void)z8;

  TDM_LOAD(g0_ld, g1_ld, z4, z8);                   // global -> LDS (TENSORcnt++)
  __builtin_amdgcn_s_wait_tensorcnt(0);             // tile resident in LDS
  TDM_STORE(g0_st, g1_st, z4, z8);                  // LDS -> global
  __builtin_amdgcn_s_wait_tensorcnt(0);
}
#endif

// ---- Fallback / reference path: coalesced float4 crop copy -----------------
// grid = (25, 2560), block = 256: blockIdx.y encodes (t, bc, z); the 6400
// threads of a z-slice cover 160 rows * 40 float4 each. b128 loads/stores.
__global__ __launch_bounds__(256) void crop_copy_f4(const float* __restrict__ data,
                                                    const float* __restrict__ seg,
                                                    float* __restrict__ out) {
  const uint32_t i  = blockIdx.x * 256u + threadIdx.x;  // [0, 6400)
  const uint32_t y  = i / 40u;
  const uint32_t x4 = i % 40u;
  const uint32_t by = blockIdx.y;
  const uint32_t z  = by % 160u;
  const uint32_t bc = (by / 160u) % 8u;
  const uint32_t t  = by / 1280u;

  const float* srcb = (t == 0u) ? data : seg;
  const size_t src_elem =
      (((size_t)(bc * 192u + (z + 16u)) * 192u + (y + 16u)) * 192u) + 16u + (size_t)x4 * 4u;
  const size_t dst_elem = (size_t)t * (8ull * 160u * 160u * 160u) +
      (((size_t)(bc * 160u + z) * 160u + y) * 160u) + (size_t)x4 * 4u;

  const float4 v = *(const float4*)(srcb + src_elem);   // global_load_b128
  *(float4*)(out + dst_elem) = v;                       // global_store_b128
}

extern "C" void kernel_launch(void* const* d_in, const int* in_sizes, int n_in,
                              void* d_out, int out_size, void* d_ws, size_t ws_size,
                              hipStream_t stream) {
  (void)in_sizes; (void)n_in; (void)out_size; (void)d_ws; (void)ws_size;
  const float* data = (const float*)d_in[0];
  const float* seg  = (const float*)d_in[1];
  float* out = (float*)d_out;

#if HAS_TDM
  dim3 grid(5, 160, 16);   // y-chunks, z-slices, (tensor, batch*channel)
  tdm_crop_copy<<<grid, 32, 0, stream>>>(data, seg, out);
#else
  dim3 grid(25, 2560);
  crop_copy_f4<<<grid, 256, 0, stream>>>(data, seg, out);
#endif
}